// HabanaOptimizerSparseAdagrad_74904229642567
// MI455X (gfx1250) — compile-verified
//
#include <hip/hip_runtime.h>
#include <stdint.h>

// Sparse Adagrad for MI455X (gfx1250, wave32).
// Memory-bound: ~0.7 GB traffic -> ~30us at 23.3 TB/s. Copy phase (512 MB)
// dominates, so it is driven by the CDNA5 Tensor Data Mover (2 instructions
// per 64KB tile, DMA engine does the streaming). Sparse phases use native
// no-return global_atomic_add_f32 at device scope.

typedef float f4 __attribute__((ext_vector_type(4)));

#define ADAGRAD_EPS 1e-10f
#define D 64               // feature dim
#define ROWS_PER_BLOCK 16  // 256 threads / (D/4 lanes per row)
#define TDM_TILE_ELTS 16000  // floats per TDM tile = 64000 B LDS; VD % 16000 == 0

__device__ __forceinline__ uint32_t rfl(uint32_t x) {
  return (uint32_t)__builtin_amdgcn_readfirstlane((int)x);
}

// ---------------------------------------------------------------------------
// Native f32 atomic add, no return (STOREcnt path), device scope -> RMW at L2.
// ---------------------------------------------------------------------------
__device__ __forceinline__ void atomic_add_f32_dev(float* p, float v) {
  asm volatile("global_atomic_add_f32 %0, %1, off scope:SCOPE_DEV"
               :: "v"((uint64_t)(uintptr_t)p), "v"(v)
               : "memory");
}

// ---------------------------------------------------------------------------
// Table copy via the Tensor Data Mover. One wave per block builds a 1-D D#
// (ISA 8.3/8.4 layout) in pinned SGPRs (group0 = s[36:39], group1 = s[40:47])
// and issues TENSOR_LOAD_TO_LDS + TENSOR_STORE_FROM_LDS. All descriptor
// dwords are readfirstlane'd -> guaranteed uniform SGPR values.
// ---------------------------------------------------------------------------
__global__ void __launch_bounds__(32) copy_tdm(const float* __restrict__ src,
                                               float* __restrict__ dst,
                                               long long nElts) {
  __shared__ float smem[TDM_TILE_ELTS];
  const long long tileBase = (long long)blockIdx.x * TDM_TILE_ELTS;
  if (tileBase >= nElts) return;
  const long long rem = nElts - tileBase;
  const uint32_t tile = (uint32_t)(rem < TDM_TILE_ELTS ? rem : TDM_TILE_ELTS);

  const uint32_t lds    = (uint32_t)(uintptr_t)&smem[0];
  const uint64_t ga_src = (uint64_t)(uintptr_t)(src + tileBase);
  const uint64_t ga_dst = (uint64_t)(uintptr_t)(dst + tileBase);

  // --- D# group 0 dwords: count=1 | lds_addr | global_addr[56:0] | type=2
  const uint32_t g0_0 = rfl(1u);
  const uint32_t g0_1 = rfl(lds);
  const uint32_t g0_2l = rfl((uint32_t)ga_src);
  const uint32_t g0_3l = rfl((uint32_t)((ga_src >> 32) & 0x01FFFFFFu) | (2u << 30));
  const uint32_t g0_2s = rfl((uint32_t)ga_dst);
  const uint32_t g0_3s = rfl((uint32_t)((ga_dst >> 32) & 0x01FFFFFFu) | (2u << 30));

  // --- D# group 1 dwords: 1-D tile, data_size=4B, no mask/pad/iterate.
  const uint32_t g1_0 = rfl(2u << 16);                            // data_size=2 (4B)
  const uint32_t g1_1 = rfl((tile & 0xFFFFu) << 16);              // tensor_dim0[15:0]
  const uint32_t g1_2 = rfl(((tile >> 16) & 0xFFFFu) | (1u << 16)); // dim0[31:16]|dim1=1
  const uint32_t g1_3 = rfl((tile & 0xFFFFu) << 16);              // dim1[31:16]=0|tile_dim0
  const uint32_t g1_4 = rfl(0u);                                  // tile_dim1=0, tile_dim2=0
  const uint32_t g1_5 = rfl(tile);                                // tensor_dim0_stride[31:0]
  const uint32_t g1_6 = rfl(0u);                                  // stride0[47:32]|stride1[15:0]
  const uint32_t g1_7 = rfl(0u);                                  // stride1[47:16]

  asm volatile("tensor_load_to_lds s[36:39], s[40:47]"
               :: "{s36}"(g0_0), "{s37}"(g0_1), "{s38}"(g0_2l), "{s39}"(g0_3l),
                  "{s40}"(g1_0), "{s41}"(g1_1), "{s42}"(g1_2), "{s43}"(g1_3),
                  "{s44}"(g1_4), "{s45}"(g1_5), "{s46}"(g1_6), "{s47}"(g1_7)
               : "memory");
  __builtin_amdgcn_s_wait_tensorcnt(0);   // tile landed in LDS

  asm volatile("tensor_store_from_lds s[36:39], s[40:47]"
               :: "{s36}"(g0_0), "{s37}"(g0_1), "{s38}"(g0_2s), "{s39}"(g0_3s),
                  "{s40}"(g1_0), "{s41}"(g1_1), "{s42}"(g1_2), "{s43}"(g1_3),
                  "{s44}"(g1_4), "{s45}"(g1_5), "{s46}"(g1_6), "{s47}"(g1_7)
               : "memory");
  __builtin_amdgcn_s_wait_tensorcnt(0);   // drain before exit
}

// ---------------------------------------------------------------------------
// Fallback bulk copy via per-lane async DMA (global -> LDS -> global, 128b,
// ASYNCcnt). Kept compiled as the proven alternative data path.
// ---------------------------------------------------------------------------
__global__ void __launch_bounds__(256) copy_async_b128(const f4* __restrict__ src,
                                                       f4* __restrict__ dst,
                                                       long long nChunks) {
  __shared__ f4 smem[256 * 4];
  const int tid = threadIdx.x;
  const long long T = (long long)gridDim.x * 256;
  for (long long i = (long long)blockIdx.x * 256 + tid; i < nChunks; i += 4 * T) {
#pragma unroll
    for (int k = 0; k < 4; ++k) {
      long long c = i + (long long)k * T;
      if (c < nChunks) {
        uint32_t lds = (uint32_t)(uintptr_t)&smem[k * 256 + tid];
        asm volatile("global_load_async_to_lds_b128 %0, %1, off"
                     :: "v"(lds), "v"((uint64_t)(uintptr_t)(src + c))
                     : "memory");
      }
    }
    asm volatile("s_wait_asynccnt 0x0" ::: "memory");
#pragma unroll
    for (int k = 0; k < 4; ++k) {
      long long c = i + (long long)k * T;
      if (c < nChunks) {
        uint32_t lds = (uint32_t)(uintptr_t)&smem[k * 256 + tid];
        asm volatile("global_store_async_from_lds_b128 %0, %1, off"
                     :: "v"((uint64_t)(uintptr_t)(dst + c)), "v"(lds)
                     : "memory");
      }
    }
    asm volatile("s_wait_asynccnt 0x0" ::: "memory");
  }
}

// ---------------------------------------------------------------------------
// Pass 1: m_out[idx[r]] += g[r]^2   (duplicates accumulate via atomics)
// ---------------------------------------------------------------------------
__global__ void __launch_bounds__(256) adagrad_scatter_moments(
    const float* __restrict__ grad, const int* __restrict__ idx,
    const int* __restrict__ vcp, float* __restrict__ m_out, int N) {
  const int r = blockIdx.x * ROWS_PER_BLOCK + (threadIdx.x >> 4);
  const int c = (threadIdx.x & 15) * 4;
  const int vc = *vcp;
  if (r >= vc || r >= N) return;
  f4 g = __builtin_nontemporal_load((const f4*)(grad + (long long)r * D + c));
  const long long base = (long long)idx[r] * D + c;
  atomic_add_f32_dev(m_out + base + 0, g.x * g.x);
  atomic_add_f32_dev(m_out + base + 1, g.y * g.y);
  atomic_add_f32_dev(m_out + base + 2, g.z * g.z);
  atomic_add_f32_dev(m_out + base + 3, g.w * g.w);
}

// ---------------------------------------------------------------------------
// Pass 2: w_out[idx[r]] += -lr * g / (sqrt(m_full) + eps)
// ---------------------------------------------------------------------------
__global__ void __launch_bounds__(256) adagrad_update_weights(
    const float* __restrict__ grad, const int* __restrict__ idx,
    const int* __restrict__ vcp, const float* __restrict__ lrp,
    const float* __restrict__ m_out, float* __restrict__ w_out, int N) {
  const int r = blockIdx.x * ROWS_PER_BLOCK + (threadIdx.x >> 4);
  const int c = (threadIdx.x & 15) * 4;
  const int vc = *vcp;
  if (r >= vc || r >= N) return;
  const float lr = *lrp;
  f4 g = __builtin_nontemporal_load((const f4*)(grad + (long long)r * D + c));
  const long long base = (long long)idx[r] * D + c;
  f4 m = *(const f4*)(m_out + base);
  f4 u;
  u.x = -lr * g.x / (__builtin_sqrtf(m.x) + ADAGRAD_EPS);
  u.y = -lr * g.y / (__builtin_sqrtf(m.y) + ADAGRAD_EPS);
  u.z = -lr * g.z / (__builtin_sqrtf(m.z) + ADAGRAD_EPS);
  u.w = -lr * g.w / (__builtin_sqrtf(m.w) + ADAGRAD_EPS);
  atomic_add_f32_dev(w_out + base + 0, u.x);
  atomic_add_f32_dev(w_out + base + 1, u.y);
  atomic_add_f32_dev(w_out + base + 2, u.z);
  atomic_add_f32_dev(w_out + base + 3, u.w);
}

extern "C" void kernel_launch(void* const* d_in, const int* in_sizes, int n_in,
                              void* d_out, int out_size, void* d_ws, size_t ws_size,
                              hipStream_t stream) {
  const float* grad = (const float*)d_in[0];  // [N, 64]
  const float* w    = (const float*)d_in[1];  // [V, 64]
  const float* m    = (const float*)d_in[2];  // [V, 64]
  const int*   idx  = (const int*)d_in[3];    // [N]
  const float* lrp  = (const float*)d_in[4];  // scalar
  const int*   vcp  = (const int*)d_in[5];    // scalar (low 32 bits)

  const long long VD = in_sizes[1];           // V*D = 32,000,000
  const int       N  = in_sizes[3];           // 262,144

  float* w_out = (float*)d_out;               // tuple order: weights, moments
  float* m_out = w_out + VD;

  const int tdmGrid = (int)((VD + TDM_TILE_ELTS - 1) / TDM_TILE_ELTS); // 2000
  copy_tdm<<<tdmGrid, 32, 0, stream>>>(w, w_out, VD);
  copy_tdm<<<tdmGrid, 32, 0, stream>>>(m, m_out, VD);

  const int gridRows = (N + ROWS_PER_BLOCK - 1) / ROWS_PER_BLOCK;
  adagrad_scatter_moments<<<gridRows, 256, 0, stream>>>(grad, idx, vcp, m_out, N);
  adagrad_update_weights<<<gridRows, 256, 0, stream>>>(grad, idx, vcp, lrp,
                                                       m_out, w_out, N);
}